// AdaptiveLowPassFilter_111669149829
// MI455X (gfx1250) — compile-verified
//
#include <hip/hip_runtime.h>
#include <hip/hip_bf16.h>

typedef __attribute__((ext_vector_type(2))) float v2f;
typedef __attribute__((ext_vector_type(8))) float v8f;

#define BB  8
#define CC  96
#define CH  48   // C/2
#define HH  128
#define WW  128
#define NP  128  // pixels (one row) per block
#define NT  256  // threads per block (8 wave32 waves)

// Manual LDS layout (floats) inside ONE flat array so the backend cannot
// reorder: weights live at offset 0 => all GEMM B-fragment ds_loads are
// base + 16-bit immediate (max byte offset 18416 < 65536).
#define W1OFF 0                      // [CH*CC]  = 4608 floats (18 KB)
#define W2OFF (W1OFF + CH * CC)      // [16*CH]  =  768 floats ( 3 KB)
#define H2OFF (W2OFF + 16 * CH)      // [NP*CH]  = 6144 floats (24 KB)
#define HOFF  (H2OFF + NP * CH)      // [NP*CC]  = 12288 floats (48 KB); reused for kw
#define SMEM_FLOATS (HOFF + NP * CC) // 23808 floats = 93 KB

__launch_bounds__(NT)
__global__ void alpf_fused_kernel(const float* __restrict__ x,
                                  const float* __restrict__ dw_w,
                                  const float* __restrict__ dw_b,
                                  const float* __restrict__ pw1_w,
                                  const float* __restrict__ pw1_b,
                                  const float* __restrict__ pw2_w,
                                  const float* __restrict__ pw2_b,
                                  float* __restrict__ out) {
    __shared__ __attribute__((aligned(16))) float smem[SMEM_FLOATS];
    float* const w1_lds = smem + W1OFF;
    float* const w2_lds = smem + W2OFF;
    float* const h2_lds = smem + H2OFF;
    float* const h_lds  = smem + HOFF;

    const int b    = blockIdx.x / HH;
    const int row  = blockIdx.x % HH;
    const int tid  = threadIdx.x;
    const int lane = tid & 31;
    const int wave = tid >> 5;

    // ---------------- Phase 0: stage pointwise weights into LDS -------------------
    for (int i = tid; i < CH * CC; i += NT) w1_lds[i] = pw1_w[i];
    for (int i = tid; i < 16 * CH; i += NT) w2_lds[i] = (i < 9 * CH) ? pw2_w[i] : 0.f;

    // ---------------- Phase 1: depthwise 3x3 + bias -> h_lds[p][c] ----------------
    {
        const int p  = tid & (NP - 1);       // column
        const int c0 = (tid >> 7) * (CC / 2);
        for (int cc = 0; cc < CC / 2; ++cc) {
            const int c = c0 + cc;
            const float* xc = x + (((size_t)b * CC + c) * HH) * WW;
            const float* wk = dw_w + c * 9;
            float acc = dw_b[c];
#pragma unroll
            for (int di = -1; di <= 1; ++di) {
                const int rr = row + di;
                if (rr < 0 || rr >= HH) continue;
#pragma unroll
                for (int dj = -1; dj <= 1; ++dj) {
                    const int cl = p + dj;
                    if (cl < 0 || cl >= WW) continue;
                    acc += xc[rr * WW + cl] * wk[(di + 1) * 3 + (dj + 1)];
                }
            }
            h_lds[p * CC + c] = acc;
        }
    }
    __syncthreads();   // covers phase 0 + phase 1

    // ---------------- Phase 2: pw1 GEMM (M=128,K=96,N=48) via v_wmma_f32_16x16x4_f32
    // One M-tile per wave; 3 independent N-accumulators share each A fragment.
    // Fused bias + LeakyReLU(0.2) -> h2_lds[p][n]
    {
        const int ml    = lane & 15;
        const int khalf = (lane >> 4) * 2;     // lanes 16-31 hold K+2,K+3
        const int m     = wave * 16 + ml;      // mt = wave (8 tiles / 8 waves)
        const float* arow = &h_lds[m * CC];
        const float* brow = &w1_lds[ml * CC];  // single base; nt rows via imm offsets
        v8f acc[3] = {v8f{}, v8f{}, v8f{}};
#pragma unroll
        for (int k0 = 0; k0 < CC; k0 += 4) {
            const int ka = k0 + khalf;         // even -> 8B aligned
            v2f a  = *(const v2f*)(arow + ka);
            v2f b0 = *(const v2f*)(brow + ka);
            v2f b1 = *(const v2f*)(brow + 16 * CC + ka);
            v2f b2 = *(const v2f*)(brow + 32 * CC + ka);
            acc[0] = __builtin_amdgcn_wmma_f32_16x16x4_f32(
                         false, a, false, b0, (short)0, acc[0], false, false);
            acc[1] = __builtin_amdgcn_wmma_f32_16x16x4_f32(
                         false, a, false, b1, (short)0, acc[1], false, false);
            acc[2] = __builtin_amdgcn_wmma_f32_16x16x4_f32(
                         false, a, false, b2, (short)0, acc[2], false, false);
        }
#pragma unroll
        for (int nt = 0; nt < 3; ++nt) {
            const int no     = nt * 16 + ml;
            const float bias = pw1_b[no];
#pragma unroll
            for (int r = 0; r < 8; ++r) {
                const int mo = wave * 16 + r + ((lane >= 16) ? 8 : 0);
                float v = acc[nt][r] + bias;
                v = (v >= 0.f) ? v : 0.2f * v;   // LeakyReLU(0.2)
                h2_lds[mo * CH + no] = v;
            }
        }
    }
    __syncthreads();

    // ---------------- Phase 3: pw2 GEMM (M=128,K=48,N=16 pad of 9) -> logits ------
    float* kw_lds = h_lds;  // reuse (h consumed)
    {
        const int ml    = lane & 15;
        const int khalf = (lane >> 4) * 2;
        const int m     = wave * 16 + ml;      // 8 M-tiles over 8 waves
        const float* arow = &h2_lds[m * CH];
        const float* brow = &w2_lds[ml * CH];
        v8f acc = {};
#pragma unroll
        for (int k0 = 0; k0 < CH; k0 += 4) {
            const int ka = k0 + khalf;
            v2f a  = *(const v2f*)(arow + ka);
            v2f bf = *(const v2f*)(brow + ka);
            acc = __builtin_amdgcn_wmma_f32_16x16x4_f32(
                      false, a, false, bf, (short)0, acc, false, false);
        }
        const int no = lane & 15;
#pragma unroll
        for (int r = 0; r < 8; ++r) {
            const int mo = wave * 16 + r + ((lane >= 16) ? 8 : 0);
            if (no < 9) kw_lds[mo * 16 + no] = acc[r] + pw2_b[no];
        }
    }
    __syncthreads();

    // ---------------- Phase 4: softmax over 9 logits per pixel --------------------
    if (tid < NP) {
        float* kp = kw_lds + tid * 16;
        float mx = kp[0];
#pragma unroll
        for (int k = 1; k < 9; ++k) mx = fmaxf(mx, kp[k]);
        float e[9], s = 0.f;
#pragma unroll
        for (int k = 0; k < 9; ++k) { e[k] = __expf(kp[k] - mx); s += e[k]; }
        const float inv = 1.f / s;
#pragma unroll
        for (int k = 0; k < 9; ++k) kp[k] = e[k] * inv;
    }
    __syncthreads();

    // ---------------- Phase 5: per-pixel 3x3 weighted average of x ----------------
    {
        const int col = tid & (NP - 1);
        const int c0  = (tid >> 7) * (CC / 2);
        float wv[9];
#pragma unroll
        for (int k = 0; k < 9; ++k) wv[k] = kw_lds[col * 16 + k];
        for (int cc = 0; cc < CC / 2; ++cc) {
            const int c = c0 + cc;
            const float* xc = x + (((size_t)b * CC + c) * HH) * WW;
            float acc = 0.f;
#pragma unroll
            for (int di = -1; di <= 1; ++di) {
                const int rr = row + di;
                if (rr < 0 || rr >= HH) continue;
#pragma unroll
                for (int dj = -1; dj <= 1; ++dj) {
                    const int cl = col + dj;
                    if (cl < 0 || cl >= WW) continue;
                    acc += xc[rr * WW + cl] * wv[(di + 1) * 3 + (dj + 1)];
                }
            }
            out[(((size_t)b * CC + c) * HH + row) * WW + col] = acc;
        }
    }
}

extern "C" void kernel_launch(void* const* d_in, const int* in_sizes, int n_in,
                              void* d_out, int out_size, void* d_ws, size_t ws_size,
                              hipStream_t stream) {
    const float* x     = (const float*)d_in[0];
    const float* dw_w  = (const float*)d_in[1];
    const float* dw_b  = (const float*)d_in[2];
    const float* pw1_w = (const float*)d_in[3];
    const float* pw1_b = (const float*)d_in[4];
    const float* pw2_w = (const float*)d_in[5];
    const float* pw2_b = (const float*)d_in[6];
    float* out = (float*)d_out;

    dim3 grid(BB * HH);   // one image row per block
    dim3 block(NT);       // 8 wave32 waves
    alpf_fused_kernel<<<grid, block, 0, stream>>>(x, dw_w, dw_b, pw1_w, pw1_b,
                                                  pw2_w, pw2_b, out);
}